// DifferentiableVoxelGrid_1374389534992
// MI455X (gfx1250) — compile-verified
//
#include <hip/hip_runtime.h>

// ---------------------------------------------------------------------------
// DifferentiableVoxelGrid pruning + material softmax for MI455X (gfx1250).
//
// Pass structure (all on `stream`, graph-capture safe):
//   0. hipMemsetAsync          : zero histograms / counters / tie counts
//   1. k_depth   (WMMA f32)    : per-voxel clip + view_z via V_WMMA_F32_16X16X4_F32,
//                                frustum+occupancy validity, depth bits, 12-bit LDS hist
//   2. k_select(level 0)       : radix-select bin over top 12 bits
//   3. k_refine(level 1)       : 12-bit refinement histogram (bits 19:8)
//   4. k_select(level 1)
//   5. k_refine(level 2)       : 8-bit refinement histogram (bits 7:0)
//   6. k_select(level 2)       : exact threshold T + #ties-to-keep r
//   7. k_tiecount              : per-block count of depth==T
//   8. k_scan                  : exclusive scan of block tie counts (index order)
//   9. k_output                : keep = db<T || (db==T && tieRank<r);
//                                out = keep ? sigmoid(occ)*softmax(mat) : 0
// ---------------------------------------------------------------------------

#define GX 192
#define GY 96
#define GZ 192
#define NVOX (GX * GY * GZ)          // 3,538,944 (divisible by 256)
#define YZ (GY * GZ)
#define NBLK (NVOX / 256)            // 13,824
#define WORLD_SCALE 2.0f
#define SENTINEL 0xFFFFFFFFu
// sigmoid(x) > 0.01  <=>  x > ln(0.01/0.99)  (sigmoid strictly monotone)
#define OCC_LOGIT_THRESHOLD (-4.5951199f)

typedef float v2f __attribute__((ext_vector_type(2)));
typedef float v8f __attribute__((ext_vector_type(8)));

// counters layout (u32): [0]=validCount [1]=prefix [2]=krem [3]=done [4]=T [5]=r

// ---------------------------------------------------------------------------
// Kernel 1: WMMA transform + validity + depth bits + level-1 histogram.
//
// D = A(16x4) x B(4x16).  A rows = extended transform rows
//   rows 0..3 : MVP = proj @ view      (clip = MVP @ [c;1])
//   row  4    : view row 2             (view_z = view[2,:] @ [c;1])
// B columns = homogeneous voxel centers [cx,cy,cz,1].
// WMMA #0 puts transform in A rows 0..4  -> lanes 0..15 own voxels base+0..15
// WMMA #1 puts transform in A rows 8..12 -> lanes 16..31 own voxels base+16..31
// so lane L owns voxel (waveBase+L): comp j = (L<16 ? d0[j] : d1[j]).
//
// Frustum test is divide-free: w = max(clip_w, 1e-6) > 0, so
//   -1 <= clip_j/w <= 1  <=>  |clip_j| <= w.
// ---------------------------------------------------------------------------
__global__ __launch_bounds__(256) void k_depth(
    const float* __restrict__ occ_logits,
    const float* __restrict__ view,   // 4x4 row-major
    const float* __restrict__ proj,   // 4x4 row-major
    unsigned* __restrict__ depthbits,
    unsigned* __restrict__ hist1,
    unsigned* __restrict__ counters)
{
  __shared__ unsigned lhist[4096];
  __shared__ unsigned lvalid;
  const int tid = threadIdx.x;
  for (int i = tid; i < 4096; i += 256) lhist[i] = 0u;
  if (tid == 0) lvalid = 0u;
  __syncthreads();

  const unsigned lane   = (unsigned)tid & 31u;
  const unsigned M      = lane & 15u;
  const bool     hiHalf = lane >= 16u;
  const int      k0     = hiHalf ? 2 : 0;

  // Extended transform row element: r<4 -> mvp[r][j], r==4 -> view[2][j]
  auto extRow = [&](int r, int j) -> float {
    if (r < 4) {
      float s = 0.0f;
#pragma unroll
      for (int t = 0; t < 4; ++t) s += proj[r * 4 + t] * view[t * 4 + j];
      return s;
    }
    return view[2 * 4 + j];
  };

  v2f a0 = {0.0f, 0.0f};
  v2f a1 = {0.0f, 0.0f};
  if (M < 5u)              { a0.x = extRow((int)M, k0);     a0.y = extRow((int)M, k0 + 1); }
  if (M >= 8u && M < 13u)  { a1.x = extRow((int)M - 8, k0); a1.y = extRow((int)M - 8, k0 + 1); }

  const int n        = blockIdx.x * 256 + tid;   // this thread's voxel
  const int waveBase = n & ~31;

  auto makeB = [&](int vbase) -> v2f {
    int v   = vbase + (int)(lane & 15u);
    int ix  = v / YZ;
    int rem = v - ix * YZ;
    int iy  = rem / GZ;
    int iz  = rem - iy * GZ;
    float cx = ((float)ix + 0.5f - (float)GX * 0.5f) * WORLD_SCALE;
    float cy = ((float)iy + 0.5f) * WORLD_SCALE;
    float cz = ((float)iz + 0.5f - (float)GZ * 0.5f) * WORLD_SCALE;
    v2f b;
    if (!hiHalf) { b.x = cx; b.y = cy; }      // K=0,1 rows
    else         { b.x = cz; b.y = 1.0f; }    // K=2,3 rows
    return b;
  };

  v2f b0 = makeB(waveBase);
  v2f b1 = makeB(waveBase + 16);
  v8f cz8 = {};

  // EXEC is full here (exact grid, structured control flow) as WMMA requires.
  v8f d0 = __builtin_amdgcn_wmma_f32_16x16x4_f32(false, a0, false, b0,
                                                 (short)0, cz8, false, false);
  v8f d1 = __builtin_amdgcn_wmma_f32_16x16x4_f32(false, a1, false, b1,
                                                 (short)0, cz8, false, false);

  const float clipx = hiHalf ? d1[0] : d0[0];
  const float clipy = hiHalf ? d1[1] : d0[1];
  const float clipz = hiHalf ? d1[2] : d0[2];
  const float clipw = hiHalf ? d1[3] : d0[3];
  const float viewz = hiHalf ? d1[4] : d0[4];

  const float w = fmaxf(clipw, 1e-6f);        // > 0
  const bool visible = (fabsf(clipx) <= w) &
                       (fabsf(clipy) <= w) &
                       (fabsf(clipz) <= w);

  const bool active = occ_logits[n] > OCC_LOGIT_THRESHOLD;
  const bool valid  = active & visible;

  const float depth = fmaxf(-viewz, 0.0f);    // >= 0, so bits are monotone
  unsigned db = SENTINEL;
  if (valid) db = __float_as_uint(depth);
  depthbits[n] = db;

  if (valid) {
    atomicAdd(&lhist[db >> 20], 1u);
    atomicAdd(&lvalid, 1u);
  }
  __syncthreads();
  for (int i = tid; i < 4096; i += 256)
    if (lhist[i]) atomicAdd(&hist1[i], lhist[i]);
  if (tid == 0 && lvalid) atomicAdd(&counters[0], lvalid);
}

// ---------------------------------------------------------------------------
// Radix-select step: find the bin containing rank k; update prefix/krem.
// ---------------------------------------------------------------------------
__global__ __launch_bounds__(256) void k_select(
    const unsigned* __restrict__ hist, int nbins, int level,
    unsigned* __restrict__ counters, const int* __restrict__ maxblocks_p)
{
  __shared__ unsigned csum[256];
  const int tid   = threadIdx.x;
  const int chunk = nbins >> 8;   // 4096 -> 16, 256 -> 1
  unsigned local = 0u;
  for (int i = 0; i < chunk; ++i) local += hist[tid * chunk + i];
  csum[tid] = local;
  __syncthreads();
  if (tid != 0) return;
  if (level > 0 && counters[3]) return;   // already resolved

  unsigned k;
  if (level == 0) {
    const unsigned vc = counters[0];
    long mk = (long)maxblocks_p[0];
    if (mk <= 0)            { counters[3] = 1u; counters[4] = 0u;       counters[5] = 0u; return; }
    if (vc <= (unsigned)mk) { counters[3] = 1u; counters[4] = SENTINEL; counters[5] = 0u; return; }
    k = (unsigned)mk;
  } else {
    k = counters[2];
  }

  unsigned run = 0u;
  int c = 0;
  for (; c < 256; ++c) { unsigned h = csum[c]; if (run + h >= k) break; run += h; }
  if (c == 256) c = 255;                   // safety (cannot trigger by construction)
  int b = c * chunk;
  for (int i = 0; i < chunk - 1; ++i) {
    unsigned h = hist[b];
    if (run + h >= k) break;
    run += h; ++b;
  }
  const unsigned krem = k - run;           // 1-indexed rank within bin b
  if (level == 0)      { counters[1] = (unsigned)b;                       counters[2] = krem; }
  else if (level == 1) { counters[1] = (counters[1] << 12) | (unsigned)b; counters[2] = krem; }
  else                 { counters[4] = (counters[1] << 8)  | (unsigned)b; counters[5] = krem;
                         counters[3] = 1u; }
}

// ---------------------------------------------------------------------------
// Refinement histograms over the selected prefix.
// ---------------------------------------------------------------------------
__global__ __launch_bounds__(256) void k_refine(
    const unsigned* __restrict__ depthbits, unsigned* __restrict__ histOut,
    const unsigned* __restrict__ counters, int level)
{
  if (counters[3]) return;                 // resolved early (keep-all / keep-none)
  const int n = blockIdx.x * 256 + threadIdx.x;
  const unsigned db = depthbits[n];
  if (db == SENTINEL) return;
  const unsigned prefix = counters[1];
  if (level == 1) {
    if ((db >> 20) == prefix) atomicAdd(&histOut[(db >> 8) & 0xFFFu], 1u);
  } else {
    if ((db >> 8) == prefix)  atomicAdd(&histOut[db & 0xFFu], 1u);
  }
}

// ---------------------------------------------------------------------------
// Per-block tie counts (depth bits == T), for index-ordered tie-breaking.
// ---------------------------------------------------------------------------
__global__ __launch_bounds__(256) void k_tiecount(
    const unsigned* __restrict__ depthbits,
    const unsigned* __restrict__ counters,
    unsigned* __restrict__ blockTie)
{
  __shared__ unsigned cnt;
  if (threadIdx.x == 0) cnt = 0u;
  __syncthreads();
  const unsigned T = counters[4];
  const int n = blockIdx.x * 256 + threadIdx.x;
  if (depthbits[n] == T) atomicAdd(&cnt, 1u);
  __syncthreads();
  if (threadIdx.x == 0) blockTie[blockIdx.x] = cnt;
}

// ---------------------------------------------------------------------------
// Single-block exclusive scan of NBLK (=13824) block tie counts.
// ---------------------------------------------------------------------------
__global__ __launch_bounds__(1024) void k_scan(
    const unsigned* __restrict__ in, unsigned* __restrict__ out, int nb)
{
  __shared__ unsigned part[1024];
  const int CH  = 14;                       // 1024*14 = 14336 >= 13824
  const int tid = threadIdx.x;
  unsigned vals[14];
  unsigned s = 0u;
  const int base = tid * CH;
#pragma unroll
  for (int i = 0; i < CH; ++i) {
    const int idx = base + i;
    const unsigned v = (idx < nb) ? in[idx] : 0u;
    vals[i] = v; s += v;
  }
  part[tid] = s;
  __syncthreads();
  if (tid == 0) {
    unsigned run = 0u;
    for (int t = 0; t < 1024; ++t) { unsigned tmp = part[t]; part[t] = run; run += tmp; }
  }
  __syncthreads();
  unsigned run = part[tid];
#pragma unroll
  for (int i = 0; i < CH; ++i) {
    const int idx = base + i;
    if (idx < nb) out[idx] = run;
    run += vals[i];
  }
}

// ---------------------------------------------------------------------------
// Final output: keep mask + occupancy-scaled material softmax (f32x4 I/O).
// ---------------------------------------------------------------------------
__global__ __launch_bounds__(256) void k_output(
    const float* __restrict__ occ_logits,
    const float* __restrict__ mat_logits,
    const unsigned* __restrict__ depthbits,
    const unsigned* __restrict__ counters,
    const unsigned* __restrict__ blockScan,
    float* __restrict__ out)
{
  __shared__ unsigned wcnt[8];
  const int tid = threadIdx.x;
  const int n   = blockIdx.x * 256 + tid;
  const unsigned T = counters[4];
  const unsigned r = counters[5];
  const unsigned db = depthbits[n];

  const bool tie = (db == T);
  const unsigned mask = __builtin_amdgcn_ballot_w32(tie);   // wave32 ballot
  const unsigned lane = (unsigned)tid & 31u;
  const unsigned wid  = (unsigned)tid >> 5;
  if (lane == 0) wcnt[wid] = (unsigned)__popc(mask);
  __syncthreads();
  unsigned pre = 0u;
  for (unsigned w = 0; w < wid; ++w) pre += wcnt[w];
  const unsigned rank = blockScan[blockIdx.x] + pre +
                        (unsigned)__popc(mask & ((1u << lane) - 1u));

  const bool keep = (db < T) || (tie && rank < r);

  float4* o = (float4*)(out + (size_t)n * 8);
  if (!keep) {
    const float4 z = make_float4(0.f, 0.f, 0.f, 0.f);
    o[0] = z; o[1] = z;
    return;
  }

  const float occ = 1.0f / (1.0f + __expf(-occ_logits[n]));
  const float4* m = (const float4*)(mat_logits + (size_t)n * 8);
  const float4 m0 = m[0], m1 = m[1];
  const float mx = fmaxf(fmaxf(fmaxf(m0.x, m0.y), fmaxf(m0.z, m0.w)),
                         fmaxf(fmaxf(m1.x, m1.y), fmaxf(m1.z, m1.w)));
  const float e0 = __expf(m0.x - mx), e1 = __expf(m0.y - mx);
  const float e2 = __expf(m0.z - mx), e3 = __expf(m0.w - mx);
  const float e4 = __expf(m1.x - mx), e5 = __expf(m1.y - mx);
  const float e6 = __expf(m1.z - mx), e7 = __expf(m1.w - mx);
  const float sum = ((e0 + e1) + (e2 + e3)) + ((e4 + e5) + (e6 + e7));
  const float sc  = occ / sum;
  o[0] = make_float4(e0 * sc, e1 * sc, e2 * sc, e3 * sc);
  o[1] = make_float4(e4 * sc, e5 * sc, e6 * sc, e7 * sc);
}

// ---------------------------------------------------------------------------
extern "C" void kernel_launch(void* const* d_in, const int* in_sizes, int n_in,
                              void* d_out, int out_size, void* d_ws, size_t ws_size,
                              hipStream_t stream) {
  (void)in_sizes; (void)n_in; (void)out_size; (void)ws_size;

  const float* occ  = (const float*)d_in[0];   // (X,Y,Z) f32
  const float* mat  = (const float*)d_in[1];   // (X,Y,Z,8) f32
  const float* view = (const float*)d_in[2];   // (4,4) f32
  const float* proj = (const float*)d_in[3];   // (4,4) f32
  const int*   mkb  = (const int*)d_in[4];     // scalar max_blocks
  float*       out  = (float*)d_out;           // (X,Y,Z,8) f32

  // workspace layout (u32 units)
  unsigned* ws        = (unsigned*)d_ws;
  unsigned* hist1     = ws;                    // 4096
  unsigned* hist2     = ws + 4096;             // 4096
  unsigned* hist3     = ws + 8192;             // 256
  unsigned* counters  = ws + 8448;             // 16
  unsigned* blockTie  = ws + 8464;             // NBLK
  unsigned* blockScan = ws + 8464 + NBLK;      // NBLK
  unsigned* depthbits = ws + 8464 + 2 * NBLK;  // NVOX

  // zero hist1..blockTie (blockScan/depthbits are fully overwritten)
  hipMemsetAsync(ws, 0, (size_t)(8464 + NBLK) * sizeof(unsigned), stream);

  k_depth   <<<NBLK, 256, 0, stream>>>(occ, view, proj, depthbits, hist1, counters);
  k_select  <<<1,    256, 0, stream>>>(hist1, 4096, 0, counters, mkb);
  k_refine  <<<NBLK, 256, 0, stream>>>(depthbits, hist2, counters, 1);
  k_select  <<<1,    256, 0, stream>>>(hist2, 4096, 1, counters, mkb);
  k_refine  <<<NBLK, 256, 0, stream>>>(depthbits, hist3, counters, 2);
  k_select  <<<1,    256, 0, stream>>>(hist3, 256, 2, counters, mkb);
  k_tiecount<<<NBLK, 256, 0, stream>>>(depthbits, counters, blockTie);
  k_scan    <<<1,   1024, 0, stream>>>(blockTie, blockScan, NBLK);
  k_output  <<<NBLK, 256, 0, stream>>>(occ, mat, depthbits, counters, blockScan, out);
}